// NGCF_20684562498309
// MI455X (gfx1250) — compile-verified
//
#include <hip/hip_runtime.h>

typedef float v2f __attribute__((ext_vector_type(2)));
typedef float v8f __attribute__((ext_vector_type(8)));

#define NEG_SLOPE 0.01f
#define D 64
#define OUTD 256   // (N_LAYERS+1)*D

// ---------------- utility kernels ----------------

__global__ void k_zero(float* __restrict__ p, int n) {
    int i = blockIdx.x * blockDim.x + threadIdx.x;
    if (i < n) p[i] = 0.0f;
}

__global__ void k_zero4(float4* __restrict__ p, int n4) {
    int i = blockIdx.x * blockDim.x + threadIdx.x;
    if (i < n4) p[i] = make_float4(0.f, 0.f, 0.f, 0.f);
}

// deg[col[e]] += 1
__global__ void k_deg(const int* __restrict__ col, float* __restrict__ deg, int E) {
    int e = blockIdx.x * blockDim.x + threadIdx.x;
    if (e < E) atomicAdd(&deg[col[e]], 1.0f);
}

// in-place: deg -> dis = deg>0 ? rsqrt(deg) : 0
__global__ void k_dis(float* __restrict__ dis, int N) {
    int i = blockIdx.x * blockDim.x + threadIdx.x;
    if (i < N) {
        float d = dis[i];
        dis[i] = (d > 0.0f) ? rsqrtf(d) : 0.0f;
    }
}

// s[col[e]] += dis[row[e]] * dis[col[e]]   (layer-independent)
__global__ void k_s(const int* __restrict__ row, const int* __restrict__ col,
                    const float* __restrict__ dis, float* __restrict__ s, int E) {
    int e = blockIdx.x * blockDim.x + threadIdx.x;
    if (e >= E) return;
    float w = dis[row[e]] * dis[col[e]];
    if (w != 0.0f) atomicAdd(&s[col[e]], w);
}

// copy concat(user_emb, item_emb) into out[:, 0:64]
__global__ void k_init(const float4* __restrict__ ue, const float4* __restrict__ ie,
                       float* __restrict__ out, int n_users, int N) {
    int i = blockIdx.x * blockDim.x + threadIdx.x;   // N*16 float4 slots
    if (i >= N * 16) return;
    int n = i >> 4, q = i & 15;
    float4 v = (n < n_users) ? ue[n * 16 + q] : ie[(size_t)(n - n_users) * 16 + q];
    *(float4*)(out + (size_t)n * OUTD + q * 4) = v;
}

// SpMM: Acc[col[e]] += norm_e * x_l[row[e]]   (16 threads per edge, 4 floats each)
__global__ void k_spmm(const int* __restrict__ row, const int* __restrict__ col,
                       const float* __restrict__ dis, const float* xbuf,
                       float* __restrict__ Acc, int E, int l) {
    int idx = blockIdx.x * blockDim.x + threadIdx.x;
    int e = idx >> 4;
    if (e >= E) return;
    int q = (idx & 15) * 4;
    int r = row[e], c = col[e];
    float w = dis[r] * dis[c];
    if (w == 0.0f) return;
    float4 xv = *(const float4*)(xbuf + (size_t)r * OUTD + l * D + q);
    float* ap = Acc + (size_t)c * D + q;
    atomicAdd(ap + 0, w * xv.x);
    atomicAdd(ap + 1, w * xv.y);
    atomicAdd(ap + 2, w * xv.z);
    atomicAdd(ap + 3, w * xv.w);
}

// Fused dense stage per layer, one wave per 16x16 output tile:
//   out[:, (l+1)*64 + :] = leakyReLU( Acc @ W1^T + (x_l (*) Acc) @ W2^T + s*(b1+b2) )
// Uses full-precision V_WMMA_F32_16X16X4_F32.
__global__ void k_gemm(const float* __restrict__ Acc,
                       const float* __restrict__ W1, const float* __restrict__ b1,
                       const float* __restrict__ W2, const float* __restrict__ b2,
                       const float* __restrict__ s,
                       float* xbuf,   // out buffer: read cols l*64.., write cols (l+1)*64..
                       int N, int l, int ntiles) {
    int wave = (blockIdx.x * blockDim.x + threadIdx.x) >> 5;   // uniform per wave
    if (wave >= ntiles) return;
    int lane = threadIdx.x & 31;
    int lo = lane & 15;        // A: M index / B: N index / C: col
    int hi = lane >> 4;        // K half selector
    int r0 = (wave >> 2) * 16;
    int o0 = (wave & 3) * 16;

    int arow = r0 + lo;
    if (arow >= N) arow = N - 1;                 // clamp (results discarded)
    const float* aptr = Acc  + (size_t)arow * D + 2 * hi;
    const float* xptr = xbuf + (size_t)arow * OUTD + l * D + 2 * hi;
    const float* w1p  = W1 + (size_t)l * D * D + (size_t)(o0 + lo) * D + 2 * hi;  // B1[k][n] = W1[n][k]
    const float* w2p  = W2 + (size_t)l * D * D + (size_t)(o0 + lo) * D + 2 * hi;

    v8f acc = {};
#pragma unroll
    for (int kk = 0; kk < D; kk += 4) {
        v2f a   = *(const v2f*)(aptr + kk);
        v2f xv  = *(const v2f*)(xptr + kk);
        v2f ax  = a * xv;
        v2f bb1 = *(const v2f*)(w1p + kk);
        v2f bb2 = *(const v2f*)(w2p + kk);
        acc = __builtin_amdgcn_wmma_f32_16x16x4_f32(false, a,  false, bb1, (short)0, acc, false, false);
        acc = __builtin_amdgcn_wmma_f32_16x16x4_f32(false, ax, false, bb2, (short)0, acc, false, false);
    }

    int colo = o0 + lo;
    float bsum = b1[l * D + colo] + b2[l * D + colo];
#pragma unroll
    for (int j = 0; j < 8; ++j) {
        int rrow = r0 + j + 8 * hi;
        int rcl  = (rrow < N) ? rrow : (N - 1);
        float v = acc[j] + s[rcl] * bsum;
        v = (v > 0.0f) ? v : NEG_SLOPE * v;
        if (rrow < N) xbuf[(size_t)rrow * OUTD + (l + 1) * D + colo] = v;
    }
}

// ---------------- host launcher ----------------

extern "C" void kernel_launch(void* const* d_in, const int* in_sizes, int n_in,
                              void* d_out, int out_size, void* d_ws, size_t ws_size,
                              hipStream_t stream) {
    const int E       = in_sizes[0] / 2;
    const int n_users = in_sizes[1] / D;
    const int n_items = in_sizes[2] / D;
    const int N       = n_users + n_items;

    const int*   ei  = (const int*)d_in[0];
    const int*   row = ei;          // edge_index[0] = source (x_j)
    const int*   col = ei + E;      // edge_index[1] = target (x_i)
    const float* ue  = (const float*)d_in[1];
    const float* ie  = (const float*)d_in[2];
    const float* W1  = (const float*)d_in[3];
    const float* b1  = (const float*)d_in[4];
    const float* W2  = (const float*)d_in[5];
    const float* b2  = (const float*)d_in[6];
    float* out = (float*)d_out;

    float* dis = (float*)d_ws;        // N floats (first used as degree)
    float* s   = dis + N;             // N floats
    float* Acc = s + N;               // N*64 floats

    const int B = 256;
    auto cdiv = [](long a, long b) { return (int)((a + b - 1) / b); };

    // graph preprocessing (once per call)
    k_zero<<<cdiv(2 * (long)N, B), B, 0, stream>>>(dis, 2 * N);           // zero dis(as deg) + s
    k_deg<<<cdiv(E, B), B, 0, stream>>>(col, dis, E);
    k_dis<<<cdiv(N, B), B, 0, stream>>>(dis, N);
    k_s<<<cdiv(E, B), B, 0, stream>>>(row, col, dis, s, E);
    k_init<<<cdiv((long)N * 16, B), B, 0, stream>>>((const float4*)ue, (const float4*)ie,
                                                    out, n_users, N);

    const int ntiles = ((N + 15) / 16) * 4;     // 16-row x 16-col output tiles
    for (int l = 0; l < 3; ++l) {
        k_zero4<<<cdiv((long)N * 16, B), B, 0, stream>>>((float4*)Acc, N * 16);
        k_spmm<<<cdiv(16 * (long)E, B), B, 0, stream>>>(row, col, dis, out, Acc, E, l);
        k_gemm<<<cdiv((long)ntiles * 32, B), B, 0, stream>>>(Acc, W1, b1, W2, b2, s,
                                                             out, N, l, ntiles);
    }
}